// Projection_41188736369073
// MI455X (gfx1250) — compile-verified
//
#include <hip/hip_runtime.h>
#include <hip/hip_bf16.h>

// CDNA5 / gfx1250, wave32. Pillar projection:
//   feat = relu(X @ W^T + b)        X:(NPTS,64) f32, W:(64,64) f32
//   out[idx] = mean of feat rows with idx = c0*R*R + c1*R + c2  (R=128)
// GEMM done with V_WMMA_F32_16X16X4_F32 (the f32 WMMA shape on CDNA5),
// scatter via global_atomic_add_f32, counts in workspace, finalize divides.

#define R_DIM 128

typedef __attribute__((ext_vector_type(2))) float v2f;
typedef __attribute__((ext_vector_type(8))) float v8f;

__global__ void zero_kernel(float* a, int na, float* b, int nb) {
    int i = blockIdx.x * blockDim.x + threadIdx.x;
    int stride = gridDim.x * blockDim.x;
    for (int j = i; j < na; j += stride) a[j] = 0.0f;
    for (int j = i; j < nb; j += stride) b[j] = 0.0f;
}

__device__ __forceinline__ void proj_axes(int pa, int& ax0, int& ax1, int& ax2) {
    ax0 = (pa == 0) ? 1 : 0;
    ax1 = (pa <= 1) ? 2 : 1;
    ax2 = (pa == 3) ? 2 : 3;
}

// One wave processes tiles of 16 points x 64 output channels.
// A (16x4 f32): lane<16 -> x[row, k..k+1]; lane>=16 -> x[row, k+2..k+3]
// B (4x16 f32): lane<16 -> Wt[k..k+1, n]; lane>=16 -> Wt[k+2..k+3, n]
// C/D (16x16 f32): vgpr r, lane L -> (M = r + 8*(L>>4), N = L&15)
__global__ __launch_bounds__(256)
void pillar_wmma_kernel(const float* __restrict__ X,
                        const float* __restrict__ W,
                        const float* __restrict__ bias,
                        const int*  __restrict__ coords,
                        const int*  __restrict__ proj_axis_p,
                        float* __restrict__ outsum,
                        float* __restrict__ cnt,
                        int nfull) {
    __shared__ float ldsW[64 * 65];   // Wt[k][n] = W[n][k], stride 65 (bank-conflict pad)

    const int tid = threadIdx.x;
    for (int i = tid; i < 64 * 64; i += 256) {
        int n = i >> 6;         // output channel (row of W)
        int k = i & 63;         // input channel
        ldsW[k * 65 + n] = W[i];
    }
    __syncthreads();

    const int lane = tid & 31;
    const int wv   = tid >> 5;
    const int nlow = lane & 15;
    const int hi   = lane >> 4;       // 0 or 1 (half-wave)
    const int khi  = hi << 1;         // 0 or 2: K sub-offset for this half

    int pa = *proj_axis_p;
    int ax0, ax1, ax2;
    proj_axes(pa, ax0, ax1, ax2);

    float bval[4];
#pragma unroll
    for (int t = 0; t < 4; ++t) bval[t] = bias[t * 16 + nlow];

    const int wave_id = blockIdx.x * 8 + wv;
    const int wstride = gridDim.x * 8;

    for (int tt = wave_id; tt < nfull; tt += wstride) {
        const int  base = tt * 16;
        const long arow = (long)(base + nlow);

        // speculative prefetch of this wave's next tile rows
        __builtin_prefetch(X + (long)(base + wstride * 16 + nlow) * 64, 0, 1);

        v8f acc[4];
#pragma unroll
        for (int t = 0; t < 4; ++t) {
            v8f c;
#pragma unroll
            for (int e = 0; e < 8; ++e) c[e] = bval[t];
            acc[t] = c;
        }

#pragma unroll
        for (int kk = 0; kk < 16; ++kk) {
            const int kA = kk * 4 + khi;
            v2f a = *(const v2f*)(X + arow * 64 + kA);
#pragma unroll
            for (int t = 0; t < 4; ++t) {
                v2f b;
                b.x = ldsW[kA * 65 + t * 16 + nlow];
                b.y = ldsW[(kA + 1) * 65 + t * 16 + nlow];
                acc[t] = __builtin_amdgcn_wmma_f32_16x16x4_f32(
                    /*neg_a=*/false, a, /*neg_b=*/false, b,
                    /*c_mod=*/(short)0, acc[t],
                    /*reuse_a=*/false, /*reuse_b=*/false);
            }
        }

        // scatter: lane handles rows base + hi*8 + r (r = vgpr index of C/D)
#pragma unroll
        for (int r = 0; r < 8; ++r) {
            const long row = (long)(base + hi * 8 + r);
            int c0 = coords[row * 4 + ax0];
            int c1 = coords[row * 4 + ax1];
            int c2 = coords[row * 4 + ax2];
            int idx = c0 * (R_DIM * R_DIM) + c1 * R_DIM + c2;
            if (nlow == 0) atomicAdd(&cnt[idx], 1.0f);
            float* op = outsum + (long)idx * 64;
#pragma unroll
            for (int t = 0; t < 4; ++t) {
                float v = acc[t][r];
                v = v > 0.0f ? v : 0.0f;   // relu
                atomicAdd(op + t * 16 + nlow, v);
            }
        }
    }
}

// Scalar fallback for the (npoints % 16) tail rows; one block per point.
__global__ void tail_kernel(const float* __restrict__ X,
                            const float* __restrict__ W,
                            const float* __restrict__ bias,
                            const int*  __restrict__ coords,
                            const int*  __restrict__ proj_axis_p,
                            float* __restrict__ outsum,
                            float* __restrict__ cnt,
                            int start) {
    const long p = (long)(start + (int)blockIdx.x);
    const int  o = threadIdx.x;   // 64 threads
    float s = bias[o];
    for (int c = 0; c < 64; ++c) s += X[p * 64 + c] * W[o * 64 + c];
    s = s > 0.0f ? s : 0.0f;
    int pa = *proj_axis_p;
    int ax0, ax1, ax2;
    proj_axes(pa, ax0, ax1, ax2);
    int c0 = coords[p * 4 + ax0];
    int c1 = coords[p * 4 + ax1];
    int c2 = coords[p * 4 + ax2];
    int idx = c0 * (R_DIM * R_DIM) + c1 * R_DIM + c2;
    atomicAdd(&outsum[(long)idx * 64 + o], s);
    if (o == 0) atomicAdd(&cnt[idx], 1.0f);
}

__global__ void finalize_kernel(float* __restrict__ out,
                                const float* __restrict__ cnt,
                                int n, int cout) {
    int i = blockIdx.x * blockDim.x + threadIdx.x;
    if (i < n) {
        float c = cnt[i / cout];
        c = c > 1.0f ? c : 1.0f;
        out[i] = out[i] / c;
    }
}

extern "C" void kernel_launch(void* const* d_in, const int* in_sizes, int n_in,
                              void* d_out, int out_size, void* d_ws, size_t ws_size,
                              hipStream_t stream) {
    // setup_inputs order: norm_coords, p_v_dist, W, b, coords_int, proj_axis
    const float* X      = (const float*)d_in[0];
    const float* W      = (const float*)d_in[2];
    const float* bvec   = (const float*)d_in[3];
    const int*   coords = (const int*)d_in[4];
    const int*   proj   = (const int*)d_in[5];
    float*       out    = (float*)d_out;

    const int cout    = in_sizes[3];            // 64
    const int cin     = in_sizes[2] / cout;     // 64
    const int npoints = in_sizes[0] / cin;      // 400000
    const int nseg    = out_size / cout;        // Bn*R*R = 65536

    float* cnt = (float*)d_ws;                  // nseg floats of scratch

    // zero sums (d_out) and counts every call
    int ztotal = out_size + nseg;
    int zblocks = (ztotal + 256 * 8 - 1) / (256 * 8);
    zero_kernel<<<zblocks, 256, 0, stream>>>(out, out_size, cnt, nseg);

    const int nfull = npoints / 16;
    int blocks = (nfull + 7) / 8;
    if (blocks > 1024) blocks = 1024;
    if (blocks < 1) blocks = 1;
    pillar_wmma_kernel<<<blocks, 256, 0, stream>>>(X, W, bvec, coords, proj,
                                                   out, cnt, nfull);

    const int tail = npoints & 15;
    if (tail > 0) {
        tail_kernel<<<tail, 64, 0, stream>>>(X, W, bvec, coords, proj,
                                             out, cnt, nfull * 16);
    }

    finalize_kernel<<<(out_size + 255) / 256, 256, 0, stream>>>(out, cnt, out_size, cout);
}